// GroupHadamardLayer_39685497815127
// MI455X (gfx1250) — compile-verified
//
#include <hip/hip_runtime.h>

typedef __attribute__((ext_vector_type(2))) float v2f;
typedef __attribute__((ext_vector_type(8))) float v8f;

#define CAP  6144          // >= total entry count (5115)
#define KC   512           // K per staged chunk (floats)
#define ROWP 516           // padded LDS row pitch (floats): stride%64banks=4 -> conflict-free
#define BUFF (16 * ROWP)   // one staging buffer, in floats (33024 B)

// ---------------------------------------------------------------------------
// Kernel 1: build v[f] = sum over entries e with flat_idx[e]==f of
//           w_flat[e] * diag_w[seg_ids[e]] * dense_W[seg_ids[e]]
// Deterministic: fixed scan order per chunk, fixed combine order. No atomics.
// ---------------------------------------------------------------------------
__global__ __launch_bounds__(256) void build_v_kernel(
    const int* __restrict__ flat_idx, const int* __restrict__ seg_ids,
    const float* __restrict__ w_flat, const float* __restrict__ diag_w,
    const float* __restrict__ dense_W, float* __restrict__ v, int total)
{
    __shared__ int   s_idx[CAP];
    __shared__ float s_cf[CAP];
    __shared__ float s_part[8][32];

    const int n = (total < CAP) ? total : CAP;
    for (int e = threadIdx.x; e < n; e += 256) {
        const int g = seg_ids[e];
        s_idx[e] = flat_idx[e];
        s_cf[e]  = w_flat[e] * diag_w[g] * dense_W[g];
    }
    __syncthreads();

    const int feat_local = threadIdx.x & 31;
    const int sub        = threadIdx.x >> 5;
    const int feat       = blockIdx.x * 32 + feat_local;

    const int chunk = (n + 7) >> 3;
    const int e0 = sub * chunk;
    int e1 = e0 + chunk; if (e1 > n) e1 = n;

    float acc = 0.0f;
    for (int e = e0; e < e1; ++e)
        if (s_idx[e] == feat) acc += s_cf[e];
    s_part[sub][feat_local] = acc;
    __syncthreads();

    if (sub == 0) {
        float s = 0.0f;
        #pragma unroll
        for (int w = 0; w < 8; ++w) s += s_part[w][feat_local];  // fixed order
        v[feat] = s;
    }
}

// ---------------------------------------------------------------------------
// Kernel 2: out = x @ v via V_WMMA_F32_16X16X4_F32, with the A-matrix staged
// through LDS using gfx1250 async copies (global_load_async_to_lds_b128,
// double-buffered, overlapped via ASYNCcnt) so every VMEM instruction moves
// 512 contiguous bytes. WMMA A-feed from LDS via conflict-free ds_load_b64:
//   lanes 0-15 : tile[row, k0+0..1]  (A VGPR0=K0, VGPR1=K1)
//   lanes 16-31: tile[row, k0+2..3]  (A VGPR0=K2, VGPR1=K3)
// B = v[k0..k0+3] broadcast to all 16 columns (b64 from global, L0/L2 hit).
// NOTE: inline-asm LDS offsets assume `tile` (the only __shared__ object)
// sits at LDS offset 0 — the standard allocation.
// ---------------------------------------------------------------------------
__global__ __launch_bounds__(256) void gemv_wmma_kernel(
    const float* __restrict__ x, const float* __restrict__ v,
    float* __restrict__ out, int nf)
{
    __shared__ float tile[2 * BUFF + 128];   // 2 staging buffers + partials

    const int tid     = threadIdx.x;
    const int lane    = tid & 31;
    const int wave    = tid >> 5;            // 0..7
    const int rowbase = blockIdx.x * 16;

    // ---- async staging addressing: 16 B per lane per issue, 8 issues/chunk.
    // idx = j*256 + tid -> row = idx>>7 (rows step by 2 per j), col = idx&127.
    const int r0    = tid >> 7;              // 0..1
    const int col16 = (tid & 127) << 4;      // byte offset within row
    unsigned long long gbase =
        (unsigned long long)(uintptr_t)x +
        ((unsigned long long)(rowbase + r0) * (unsigned long long)nf) * 4ull +
        (unsigned long long)col16;
    const unsigned long long grow2 = (unsigned long long)nf * 8ull;  // +2 rows
    const unsigned lds0 = (unsigned)(r0 * (ROWP * 4) + col16);

    // ---- compute addressing
    const int m   = lane & 15;
    const int hi  = lane >> 4;               // 0: K pair 0/1, 1: K pair 2/3
    const int ab0 = m * ROWP + wave * 64 + 2 * hi;   // float index in buffer
    const float* bp0 = v + wave * 64 + 2 * hi;

    const int nchunks = nf / KC;             // 8 for nf=4096
    v8f acc = {};

    // prologue: stage chunk 0 into buffer 0
    {
        unsigned long long g = gbase;
        unsigned l = lds0;
        #pragma unroll
        for (int j = 0; j < 8; ++j) {
            asm volatile("global_load_async_to_lds_b128 %0, %1, off"
                         :: "v"(l), "v"(g) : "memory");
            g += grow2; l += 2 * ROWP * 4;
        }
    }

    for (int c = 0; c < nchunks; ++c) {
        if (c + 1 < nchunks) {
            // stage chunk c+1 into the other buffer, then wait for chunk c
            unsigned long long g = gbase + (unsigned long long)(c + 1) * (KC * 4);
            unsigned l = lds0 + ((c + 1) & 1) * (BUFF * 4);
            #pragma unroll
            for (int j = 0; j < 8; ++j) {
                asm volatile("global_load_async_to_lds_b128 %0, %1, off"
                             :: "v"(l), "v"(g) : "memory");
                g += grow2; l += 2 * ROWP * 4;
            }
            asm volatile("s_wait_asynccnt 0x8" ::: "memory");  // chunk c done
        } else {
            asm volatile("s_wait_asynccnt 0x0" ::: "memory");  // last chunk done
        }
        __syncthreads();                      // chunk c visible to all waves

        const float* ap = &tile[(c & 1) * BUFF + ab0];
        const float* bp = bp0 + c * KC;
        #pragma unroll
        for (int i = 0; i < 16; ++i) {
            v2f a = *(const v2f*)ap;          // ds_load_b64, conflict-free
            v2f b = *(const v2f*)bp;          // global_load_b64 (cached)
            acc = __builtin_amdgcn_wmma_f32_16x16x4_f32(
                      false, a, false, b, (short)0, acc, false, false);
            ap += 4; bp += 4;
        }
        __syncthreads();   // done reading buf (c&1) before it is refilled
    }

    // D layout: VGPR r, lanes 0-15 -> row r; lanes 16-31 -> row r+8.
    // All 16 columns identical, so lanes 0 and 16 carry the full result.
    float* part = &tile[2 * BUFF];
    if ((lane & 15) == 0) {
        #pragma unroll
        for (int r = 0; r < 8; ++r) part[wave * 16 + hi * 8 + r] = acc[r];
    }
    __syncthreads();

    if (tid < 16) {
        float s = 0.0f;
        #pragma unroll
        for (int w = 0; w < 8; ++w) s += part[w * 16 + tid];  // fixed order
        out[rowbase + tid] = s;
    }
}

// ---------------------------------------------------------------------------
extern "C" void kernel_launch(void* const* d_in, const int* in_sizes, int n_in,
                              void* d_out, int out_size, void* d_ws, size_t ws_size,
                              hipStream_t stream) {
    const float* x        = (const float*)d_in[0];
    const int*   flat_idx = (const int*)  d_in[1];
    const int*   seg_ids  = (const int*)  d_in[2];
    const float* w_flat   = (const float*)d_in[3];
    const float* diag_w   = (const float*)d_in[4];
    const float* dense_W  = (const float*)d_in[5];
    float*       out      = (float*)d_out;
    float*       v        = (float*)d_ws;    // nf floats of scratch

    const int total = in_sizes[1];           // 5115
    const int batch = out_size;              // 8192
    const int nf    = in_sizes[0] / batch;   // 4096

    build_v_kernel<<<nf / 32, 256, 0, stream>>>(
        flat_idx, seg_ids, w_flat, diag_w, dense_W, v, total);
    gemv_wmma_kernel<<<batch / 16, 256, 0, stream>>>(x, v, out, nf);
}